// CenterOfMassLoss_266287972663
// MI455X (gfx1250) — compile-verified
//
#include <hip/hip_runtime.h>
#include <math.h>

typedef __attribute__((ext_vector_type(2))) float v2f;
typedef __attribute__((ext_vector_type(8))) float v8f;

#define NCH 31
#define IMG 2048
#define NREG 9
#define NCOMBO (NREG * NCH)          // 279
#define NSPLIT 5
#define ROWS_PER_SPLIT 80            // 400 / 5
#define PAIRS_PER_CHUNK (ROWS_PER_SPLIT * 200)  // 16000 float2-pairs per (combo,split)
#define GS_BLOCKS 1024
#define WS_REGION_BASE GS_BLOCKS     // float index into workspace
#define FUNDAMENTAL_REGION 4

// Sum-accumulator via WMMA: A = all-ones 16x4, B carries 64 data values.
// D[m][n] = sum_k B[k][n] + C[m][n]. Summing ALL 256 D slots gives
// 16 * (sum of all B elements) regardless of A/B/C register layout.
__device__ __forceinline__ v8f wmma_sum(v2f b, v8f c) {
  v2f ones; ones[0] = 1.0f; ones[1] = 1.0f;
  return __builtin_amdgcn_wmma_f32_16x16x4_f32(
      /*neg_a=*/false, ones, /*neg_b=*/false, b,
      /*c_mod=*/(short)0, c, /*reuse_a=*/false, /*reuse_b=*/false);
}

__device__ __forceinline__ float wave_red(float v) {
  for (int m = 16; m; m >>= 1) v += __shfl_xor(v, m, 32);
  return v;
}

// ---------------- Kernel 1: global sum of target (for mean_target) ----------
__global__ void __launch_bounds__(256)
k_gsum(const float4* __restrict__ t4, float* __restrict__ ws) {
  __shared__ float red[256];
  const size_t N4 = (size_t)NCH * IMG * IMG / 4;   // 32,505,856
  float s = 0.0f;
  for (size_t i = (size_t)blockIdx.x * 256 + threadIdx.x; i < N4;
       i += (size_t)GS_BLOCKS * 256) {
    float4 v = t4[i];
    s += (v.x + v.y) + (v.z + v.w);
  }
  red[threadIdx.x] = s;
  __syncthreads();
  for (int off = 128; off; off >>= 1) {
    if ((int)threadIdx.x < off) red[threadIdx.x] += red[threadIdx.x + off];
    __syncthreads();
  }
  if (threadIdx.x == 0) ws[blockIdx.x] = red[0];
}

// ---------------- Kernel 2: per (region,channel,split) reductions -----------
__global__ void __launch_bounds__(256)
k_region(const float* __restrict__ pred, const float* __restrict__ targ,
         float* __restrict__ ws) {
  const int combo  = blockIdx.x;           // region*31 + channel
  const int split  = blockIdx.y;           // 0..4, 80 rows each
  const int region = combo / NCH;
  const int chan   = combo % NCH;
  const int rb = region / 3, cb = region % 3;
  const int b1 = (rb == 0) ? 200 : ((rb == 1) ? 1000 : 1500);
  const int b3 = (cb == 0) ? 200 : ((cb == 1) ? 1000 : 1500);
  const int tid  = threadIdx.x;
  const int lane = tid & 31;
  const int wave = tid >> 5;
  const int rowOff = split * ROWS_PER_SPLIT;

  v8f aS  = {0,0,0,0,0,0,0,0};   // sum p^3
  v8f aSx = {0,0,0,0,0,0,0,0};   // sum row*p^3
  v8f aSy = {0,0,0,0,0,0,0,0};   // sum col*p^3
  v8f bS  = {0,0,0,0,0,0,0,0};   // sum t^3
  v8f bSx = {0,0,0,0,0,0,0,0};
  v8f bSy = {0,0,0,0,0,0,0,0};
  v8f cT  = {0,0,0,0,0,0,0,0};   // sum raw t (for weighting)

  const float2* p2 = (const float2*)pred;
  const float2* t2 = (const float2*)targ;

  const int niter = (PAIRS_PER_CHUNK + 255) / 256;  // 63 (uniform, no divergence)
  for (int it = 0; it < niter; ++it) {
    unsigned pi = (unsigned)(it * 256) + (unsigned)tid;
    bool valid  = pi < (unsigned)PAIRS_PER_CHUNK;
    unsigned pc = valid ? pi : 0u;                  // clamp: keep address in-bounds
    unsigned lr  = pc / 200u;                       // local row within split [0,80)
    unsigned rem = pc - lr * 200u;                  // pair index within row [0,200)
    int y = b1 + rowOff + (int)lr;
    size_t pa = ((size_t)chan * IMG + (size_t)y) * (IMG / 2) + (size_t)(b3 / 2) + rem;
    float2 p = p2[pa];
    float2 t = t2[pa];
    // branch-free masking; zero data contributes nothing to any sum
    p.x = valid ? p.x : 0.0f;  p.y = valid ? p.y : 0.0f;
    t.x = valid ? t.x : 0.0f;  t.y = valid ? t.y : 0.0f;

    float rw = (float)(rowOff + (int)lr);           // ix weight (axis 2)
    float c0 = (float)(2u * rem);                   // iy weight (axis 3)
    float c1 = c0 + 1.0f;
    float vp0 = p.x * p.x * p.x, vp1 = p.y * p.y * p.y;
    float vt0 = t.x * t.x * t.x, vt1 = t.y * t.y * t.y;

    v2f B;
    B[0] = vp0;      B[1] = vp1;       aS  = wmma_sum(B, aS);
    B[0] = rw * vp0; B[1] = rw * vp1;  aSx = wmma_sum(B, aSx);
    B[0] = c0 * vp0; B[1] = c1 * vp1;  aSy = wmma_sum(B, aSy);
    B[0] = vt0;      B[1] = vt1;       bS  = wmma_sum(B, bS);
    B[0] = rw * vt0; B[1] = rw * vt1;  bSx = wmma_sum(B, bSx);
    B[0] = c0 * vt0; B[1] = c1 * vt1;  bSy = wmma_sum(B, bSy);
    B[0] = t.x;      B[1] = t.y;       cT  = wmma_sum(B, cT);
  }

  // Collapse: sum all 8 accumulator components, cross-lane reduce, /16.
  auto collapse = [&](v8f a) -> float {
    float s = ((a[0] + a[1]) + (a[2] + a[3])) + ((a[4] + a[5]) + (a[6] + a[7]));
    return wave_red(s) * (1.0f / 16.0f);
  };
  float q0 = collapse(aS),  q1 = collapse(aSx), q2 = collapse(aSy);
  float q3 = collapse(bS),  q4 = collapse(bSx), q5 = collapse(bSy);
  float q6 = collapse(cT);

  __shared__ float red[8][7];
  if (lane == 0) {
    red[wave][0] = q0; red[wave][1] = q1; red[wave][2] = q2;
    red[wave][3] = q3; red[wave][4] = q4; red[wave][5] = q5;
    red[wave][6] = q6;
  }
  __syncthreads();
  if (tid < 7) {
    float s = 0.0f;
    for (int w = 0; w < 8; ++w) s += red[w][tid];   // fixed order: deterministic
    ws[WS_REGION_BASE + (combo * NSPLIT + split) * 7 + tid] = s;
  }
}

// ---------------- Kernel 3: epilogue (one wave) -----------------------------
__global__ void k_final(const float* __restrict__ ws, float* __restrict__ out) {
  const int lane = threadIdx.x;  // 32 threads
  float g = 0.0f;
  for (int i = lane; i < GS_BLOCKS; i += 32) g += ws[i];
  g = wave_red(g);
  const float meanT = g / (float)((size_t)NCH * IMG * IMG);

  float tot = 0.0f;
  for (int combo = lane; combo < NCOMBO; combo += 32) {
    float q[7] = {0, 0, 0, 0, 0, 0, 0};
    for (int s = 0; s < NSPLIT; ++s)
      for (int i = 0; i < 7; ++i)
        q[i] += ws[WS_REGION_BASE + (combo * NSPLIT + s) * 7 + i];
    // 0/0 -> NaN -> 0 in the reference; with x^3 >= 0, S==0 implies Sx==Sy==0.
    float cxp = (q[0] != 0.0f) ? q[1] / q[0] : 0.0f;
    float cyp = (q[0] != 0.0f) ? q[2] / q[0] : 0.0f;
    float cxt = (q[3] != 0.0f) ? q[4] / q[3] : 0.0f;
    float cyt = (q[3] != 0.0f) ? q[5] / q[3] : 0.0f;
    float dx = cxp - cxt, dy = cyp - cyt;
    float nrm = sqrtf(dx * dx + dy * dy);
    float wgt = (q[6] * (1.0f / 160000.0f)) / meanT;  // region mean / global mean
    float term = nrm * wgt;
    if (combo / NCH == FUNDAMENTAL_REGION) term *= 5.0f;  // FUNDA_WEIGHT
    tot += term;
  }
  tot = wave_red(tot);
  if (lane == 0) out[0] = tot / (float)NCOMBO;  // / (CHANNELS * NREG)
}

// ---------------- Launch ----------------------------------------------------
extern "C" void kernel_launch(void* const* d_in, const int* in_sizes, int n_in,
                              void* d_out, int out_size, void* d_ws, size_t ws_size,
                              hipStream_t stream) {
  const float* pred = (const float*)d_in[0];
  const float* targ = (const float*)d_in[1];
  float* ws  = (float*)d_ws;   // uses 1024 + 279*5*7 = 10,789 floats (~42 KB)
  float* out = (float*)d_out;

  k_gsum<<<GS_BLOCKS, 256, 0, stream>>>((const float4*)targ, ws);
  k_region<<<dim3(NCOMBO, NSPLIT), 256, 0, stream>>>(pred, targ, ws);
  k_final<<<1, 32, 0, stream>>>(ws, out);
}